// HarmonicOscillator_4638564680518
// MI455X (gfx1250) — compile-verified
//
#include <hip/hip_runtime.h>
#include <hip/hip_bf16.h>
#include <math.h>

// ---------------------------------------------------------------------------
// HarmonicOscillator for MI455X (gfx1250, wave32).
//
// Roofline: total data ~4MB (L2-resident on 192MB L2; 23.3TB/s -> ~0.2us
// floor).  Naive reference = 33M sin() evals -> transcendental-bound.
// Optimizations:
//   * phase factorization: sin(2*pi*p*phi) = sin(2*pi*p*frac(phi)) -> one
//     cumsum per batch row, kept in [0,1) for precision.
//   * Chebyshev recurrence s_{p+1} = 2c*s_p - s_{p-1} -> 2 transcendentals
//     per sample instead of 64 (32x reduction of the dominant cost).
//   * closed-form within-frame phase partials (linear upsample weights sum
//     analytically) -> no serial per-sample cumsum.
//   * Nyquist cutoff is monotone in p and wave-uniform (each wave32 = one
//     half-frame): split partial loop into mask-free fast region (float4 LDS
//     chunks), exact masked tail, and early exit past the cutoff (~3x VALU).
//   * amplitude tile staged global->LDS with CDNA5 async-tensor path
//     (global_load_async_to_lds_b32 / s_wait_asynccnt), read back as
//     same-address LDS broadcasts in the partial loop.
// ---------------------------------------------------------------------------

#define B_   4
#define P_   64
#define T_   2000
#define FL_  64
#define N_   (T_ * FL_)          // 128000 audio samples per batch row
#define SRF  16000.0f
#define NYQ  8000.0f
#define GAIN 0.02f

// ---------------------------------------------------------------------------
// Kernel 1: exclusive prefix scan of per-frame phase increments, frac-reduced.
// Frame t's exact contribution to cumsum(upsample(f0)/SR):
//   S[t] = (8*f0[t-1] + 48*f0[t] + 8*f0[t+1]) / SR   (edges clamped)
// One block per batch row; Hillis-Steele scan in LDS, double carry.
// ---------------------------------------------------------------------------
__global__ __launch_bounds__(256)
void ho_frame_scan(const float* __restrict__ f0, float* __restrict__ Pw) {
    const int b   = blockIdx.x;
    const int tid = threadIdx.x;
    __shared__ double sS[256];
    __shared__ double sCarry;
    if (tid == 0) sCarry = 0.0;
    __syncthreads();

    for (int t0 = 0; t0 < T_; t0 += 256) {
        const int t = t0 + tid;
        double S = 0.0;
        if (t < T_) {
            const int tm1 = max(t - 1, 0), tp1 = min(t + 1, T_ - 1);
            const float fm = f0[b * T_ + tm1];
            const float fc = f0[b * T_ + t];
            const float fp = f0[b * T_ + tp1];
            S = ((double)fm * 8.0 + (double)fc * 48.0 + (double)fp * 8.0) *
                (1.0 / (double)SRF);
        }
        sS[tid] = S;
        __syncthreads();
        for (int off = 1; off < 256; off <<= 1) {
            const double add = (tid >= off) ? sS[tid - off] : 0.0;
            __syncthreads();
            sS[tid] += add;
            __syncthreads();
        }
        if (t < T_) {
            double ph = sCarry + ((tid > 0) ? sS[tid - 1] : 0.0);  // exclusive
            Pw[b * T_ + t] = (float)(ph - floor(ph));              // frac [0,1)
        }
        __syncthreads();
        if (tid == 0) {
            double nc = sCarry + sS[255];
            sCarry = nc - floor(nc);
        }
        __syncthreads();
    }
}

// ---------------------------------------------------------------------------
// Kernel 2: synthesis.  Block = (batch row b, 4 consecutive frames).
// Thread = one audio sample; each wave32 = one half-frame (branch- and
// cutoff-uniform).  g = local frame (0..3), js = sample-in-frame.
// Upsample weights (align_corners=False, scale 64):
//   js < 32 : interp(f[t-1], f[t]),  w = (js+32.5)/64
//   js >= 32: interp(f[t],  f[t+1]), w = (js-31.5)/64
// Within-frame phase partial sums (closed form):
//   js < 32 : f0m*((j+1)-W1) + f0c*W1,                    W1 = (j+1)(j+65)/128
//   js >= 32: 8*f0m + 24*f0c + f0c*((u+1)-W2) + f0p*W2,   W2 = (u+1)^2/128
// ---------------------------------------------------------------------------
__global__ __launch_bounds__(256)
void ho_synth(const float* __restrict__ f0,
              const float* __restrict__ ha,   // (B,P,T)
              const float* __restrict__ amp,
              const float* __restrict__ Pw,   // frac'd exclusive frame phases
              float* __restrict__ out) {
    const int blocksPerB = T_ / 4;                 // 500
    const int b   = blockIdx.x / blocksPerB;
    const int t0  = (blockIdx.x % blocksPerB) * 4; // first frame of block
    const int tid = threadIdx.x;

    // Staged amplitude tile: frames (t0-1 .. t0+4) clamped, 64 partials each.
    __shared__ __align__(16) float s_amp[6 * 64];

    // ---- async global -> LDS staging (CDNA5 async-tensor path) ----
    const unsigned lds_base = (unsigned)(size_t)(void*)s_amp; // LDS byte offset
    for (int idx = tid; idx < 6 * 64; idx += 256) {
        const int fi = idx >> 6;           // staged frame slot 0..5
        const int p  = idx & 63;           // partial
        const int tf = min(max(t0 - 1 + fi, 0), T_ - 1);
        const float* gp = ha + ((size_t)b * P_ + p) * T_ + tf;
        const unsigned lo = lds_base + (unsigned)idx * 4u;
        asm volatile("global_load_async_to_lds_b32 %0, %1, off"
                     :: "v"(lo), "v"(gp) : "memory");
    }
    asm volatile("s_wait_asynccnt 0x0" ::: "memory");
    __syncthreads();

    const int g  = tid >> 6;       // local frame 0..3
    const int js = tid & 63;       // sample within frame
    const int t  = t0 + g;
    const int rb = b * T_;
    const int tm1 = max(t - 1, 0), tp1 = min(t + 1, T_ - 1);
    const float f0m = f0[rb + tm1];
    const float f0c = f0[rb + t];
    const float f0p = f0[rb + tp1];

    float part, wA, f0A, f0B, lA, lB;
    int   fA, fB;                  // LDS slots (t-1 -> g, t -> g+1, t+1 -> g+2)
    if (js < 32) {                 // wave-uniform branch
        const float j1 = (float)js + 1.0f;
        const float W1 = j1 * ((float)js + 65.0f) * (1.0f / 128.0f);
        part = f0m * (j1 - W1) + f0c * W1;
        wA   = ((float)js + 32.5f) * (1.0f / 64.0f);
        fA = g;     fB = g + 1;  f0A = f0m; f0B = f0c;
        lA = amp[rb + tm1];      lB = amp[rb + t];
    } else {
        const float u  = (float)(js - 32);
        const float u1 = u + 1.0f;
        const float W2 = u1 * u1 * (1.0f / 128.0f);
        part = 8.0f * f0m + 24.0f * f0c + f0c * (u1 - W2) + f0p * W2;
        wA   = (u + 0.5f) * (1.0f / 64.0f);
        fA = g + 1; fB = g + 2;  f0A = f0c; f0B = f0p;
        lA = amp[rb + t];        lB = amp[rb + tp1];
    }

    float phi = Pw[rb + t] + part * (1.0f / SRF);
    phi -= floorf(phi);                              // frac -> [0,1)
    const float th = 6.28318530717958647692f * phi;  // 2*pi*frac(phi)
    float s1, c1;
    __sincosf(th, &s1, &c1);                         // the only transcendentals

    // Wave-uniform Nyquist cutoffs (mask keeps iff f0*(p+1) < NYQ, monotone
    // in p).  cmin: safe under-estimate of kept count on BOTH frames (f0 >=
    // 100Hz gives >=100x margin over float rounding).  bound: safe
    // over-estimate of the larger cutoff; p >= bound contributes exactly 0.
    const float fmx = fmaxf(f0A, f0B), fmn = fminf(f0A, f0B);
    const int cmin  = min(64, max(0, (int)(NYQ / fmx) - 1));
    const int bound = min(64, (int)(NYQ / fmn) + 1);

    // Chebyshev: sin((p+1)*th) via s_{next} = 2c*s_cur - s_prev
    const float twoc = 2.0f * c1;
    float sprev = 0.0f, scur = s1, acc = 0.0f;
    const float* aAp = &s_amp[fA * 64];
    const float* aBp = &s_amp[fB * 64];

    int p = 0;
    // Fast region: both frames below Nyquist for all these partials -> no
    // masks; float4 LDS broadcasts (ds_load_b128).
    for (; p + 4 <= cmin; p += 4) {
        const float4 vA = *reinterpret_cast<const float4*>(&aAp[p]);
        const float4 vB = *reinterpret_cast<const float4*>(&aBp[p]);
#pragma unroll
        for (int q = 0; q < 4; ++q) {
            const float aAv = (&vA.x)[q];
            const float aBv = (&vB.x)[q];
            const float a   = aAv + wA * (aBv - aAv);
            acc = fmaf(a, scur, acc);
            const float snext = fmaf(twoc, scur, -sprev);
            sprev = scur;
            scur  = snext;
        }
    }
    // Masked tail: exact per-element Nyquist masks; stop at bound (all-zero
    // contributions beyond it).
    for (; p < bound; ++p) {
        const float m = (float)(p + 1);
        float aAv = aAp[p];
        float aBv = aBp[p];
        if (f0A * m >= NYQ) aAv = 0.0f;
        if (f0B * m >= NYQ) aBv = 0.0f;
        const float a = aAv + wA * (aBv - aAv);
        acc = fmaf(a, scur, acc);
        const float snext = fmaf(twoc, scur, -sprev);
        sprev = scur;
        scur  = snext;
    }

    const float loud = lA + wA * (lB - lA);
    const int   n    = t * FL_ + js;
    out[(size_t)b * N_ + n] = acc * loud * GAIN;
}

// ---------------------------------------------------------------------------
extern "C" void kernel_launch(void* const* d_in, const int* in_sizes, int n_in,
                              void* d_out, int out_size, void* d_ws, size_t ws_size,
                              hipStream_t stream) {
    const float* f0  = (const float*)d_in[0];   // (B,T)
    const float* ha  = (const float*)d_in[1];   // (B,P,T)
    const float* amp = (const float*)d_in[2];   // (B,T)
    float* Pw  = (float*)d_ws;                  // B*T floats = 32 KB scratch
    float* out = (float*)d_out;                 // (B, T*FL)

    ho_frame_scan<<<B_, 256, 0, stream>>>(f0, Pw);
    ho_synth<<<B_ * (T_ / 4), 256, 0, stream>>>(f0, ha, amp, Pw, out);
}